// RIM_15375982919796
// MI455X (gfx1250) — compile-verified
//
#include <hip/hip_runtime.h>
#include <hip/hip_bf16.h>

// ---- problem constants (fixed by the reference) ----
#define BATCH 8
#define CIN   768
#define NPTS  4096
#define LTOK  256
#define KNN   16
#define TC    512
#define OC    768
#define HEADS 4
#define NCHUNK 512   // projector-phase spatial chunk

#define KP    32     // GEMM K-panel staged in LDS
#define LDSP  36     // padded LDS row stride (dwords) to spread banks

typedef __attribute__((ext_vector_type(2))) float v2f;
typedef __attribute__((ext_vector_type(8))) float v8f;

// ---- explicit global-address-space accessors (force global_*, not flat_*) --
__device__ __forceinline__ v2f gload_v2f(const float* p) {
  return *(const __attribute__((address_space(1))) v2f*)(void*)p;
}
__device__ __forceinline__ float gload_f(const float* p) {
  return *(const __attribute__((address_space(1))) float*)(void*)p;
}
__device__ __forceinline__ void gstore_f(float* p, float v) {
  *(__attribute__((address_space(1))) float*)(void*)p = v;
}

// ---- async global->LDS staging (gfx1250 ASYNCcnt path), guarded ----
#if __has_builtin(__builtin_amdgcn_global_load_async_to_lds_b32)
#define HAS_ASYNC_LDS 1
__device__ __forceinline__ void async_cp_b32(const float* g, float* l) {
  __builtin_amdgcn_global_load_async_to_lds_b32(
      (__attribute__((address_space(1))) int*)(void*)g,
      (__attribute__((address_space(3))) int*)(void*)l, 0, 0);
}
#if __has_builtin(__builtin_amdgcn_s_wait_asynccnt)
#define ASYNC_WAIT_16() __builtin_amdgcn_s_wait_asynccnt(16)
#define ASYNC_WAIT_0()  __builtin_amdgcn_s_wait_asynccnt(0)
#else
#define ASYNC_WAIT_16() asm volatile("s_wait_asynccnt 16" ::: "memory")
#define ASYNC_WAIT_0()  asm volatile("s_wait_asynccnt 0" ::: "memory")
#endif
#else
// synchronous fallback: global load + ds_store; barriers provide ordering
__device__ __forceinline__ void async_cp_b32(const float* g, float* l) { *l = *g; }
#define ASYNC_WAIT_16()
#define ASYNC_WAIT_0()
#endif

// ---------------------------------------------------------------------------
// KNN gather + max-pool over K:  pooled[b,c,l] = max_k feat[b,c,idx[b,l,k]]
// ---------------------------------------------------------------------------
__global__ __launch_bounds__(256) void gather_maxpool(
    const float* __restrict__ feat, const int* __restrict__ idx,
    float* __restrict__ pooled)
{
  const int b = blockIdx.y;
  const int l = blockIdx.x;
  __shared__ int sidx[KNN];
  if (threadIdx.x < KNN)
    sidx[threadIdx.x] = idx[((long)b * LTOK + l) * KNN + threadIdx.x];
  __syncthreads();
  for (int c = threadIdx.x; c < CIN; c += 256) {
    const float* fb = feat + ((long)b * CIN + c) * NPTS;
    float m = -3.402823466e38f;
#pragma unroll
    for (int k = 0; k < KNN; ++k) m = fmaxf(m, gload_f(fb + sidx[k]));
    gstore_f(pooled + ((long)b * CIN + c) * LTOK + l, m);
  }
}

// ---------------------------------------------------------------------------
// Generic strided batched GEMM on V_WMMA_F32_16X16X4_F32.
//   D[z,m,n] = op( alpha * sum_k A[z,m,k]*B[z,k,n]  (+ R[z,m,n]) )
// A element (m,k) at A + z*sAb + m*sAm + k*sAk   (transA = swap strides)
// B element (k,n) at B + z*sBb + k*sBk + n*sBn
// R/D columns contiguous (stride 1), row strides sRm/sDm.
//
// Wave tile 64(M) x 32(N) -> 4x2 fragments, 8 accumulators.
// Block: 8 waves as 2(M) x 4(N) -> 128 x 128 tile.
// B K-panels (KP x 128) staged transposed in LDS (Bs[n][k]), double-buffered
// via async global->LDS copies; B fragment reads become ds_load_b64.
// Requires: M mult of 64, N mult of 128, K mult of KP(=32).
// AVEC: A contiguous in k (sAk==1) -> v2f fragment loads.
// ---------------------------------------------------------------------------
template <bool AVEC>
__global__ __launch_bounds__(256) void wmma_gemm_f32(
    const float* __restrict__ A, const float* __restrict__ B,
    const float* __restrict__ R, float* __restrict__ D,
    int M, int N, int K,
    long sAm, long sAk, long sAb,
    long sBk, long sBn, long sBb,
    long sRm, long sRb,
    long sDm, long sDb,
    float alpha, int relu, int hasR)
{
  __shared__ float Bs[2][128 * LDSP];

  const int tid  = threadIdx.x;
  const int w    = tid >> 5;
  const int lane = tid & 31;
  const int half = lane >> 4;   // 0: lanes 0-15, 1: lanes 16-31
  const int lm   = lane & 15;
  const int z    = blockIdx.z;

  const int row0  = blockIdx.y * 128 + (w >> 2) * 64;  // wave: 64 rows
  const int col0w = (w & 3) * 32;                      // wave col in block tile
  const int col0  = blockIdx.x * 128 + col0w;
  const bool active = (row0 < M) && (col0 < N);        // wave-uniform

  A += (long)z * sAb;
  B += (long)z * sBb;
  D += (long)z * sDb;
  if (hasR) R += (long)z * sRb;

  // ---- B staging assignment (all 256 threads, regardless of `active`) ----
  // element (kk, nn) of the panel: kk = k in panel, nn = n in block tile
  const int snn = tid & 127;
  const int skk = tid >> 7;                 // 0/1, advances by 2 per j
  const float* gB = B + (long)skk * sBk + (long)(blockIdx.x * 128 + snn) * sBn;
  const int  lbase = snn * LDSP + skk;
  const long gstep = 2 * sBk;               // per staging iteration j
  const long pstep = (long)KP * sBk;        // per panel

  // ---- A fragment pointers (strength-reduced) ----
  const float* pa[4];
#pragma unroll
  for (int mi = 0; mi < 4; ++mi)
    pa[mi] = A + (long)(row0 + mi * 16 + lm) * sAm + (long)(2 * half) * sAk;
  const long astep = 4 * sAk;
  const long apf   = 64 * sAk;              // prefetch distance (16 k-steps)

  v8f acc[4][2] = {};
  const int bco[2] = {col0w + lm, col0w + 16 + lm};        // LDS columns
  const int bc [2] = {col0 + lm, col0 + 16 + lm};          // global columns

  const int npan = K / KP;

  // stage panel 0 into Bs[0]
#pragma unroll
  for (int j = 0; j < KP / 2; ++j)
    async_cp_b32(gB + (long)j * gstep, &Bs[0][lbase + 2 * j]);
  gB += pstep;

  int bf = 0;
  for (int p = 0; p < npan; ++p) {
    if (p + 1 < npan) {
#pragma unroll
      for (int j = 0; j < KP / 2; ++j)
        async_cp_b32(gB + (long)j * gstep, &Bs[bf ^ 1][lbase + 2 * j]);
      gB += pstep;
      ASYNC_WAIT_16();                      // panel p complete, p+1 in flight
    } else {
      ASYNC_WAIT_0();
    }
    __syncthreads();                        // panel p visible to all waves

    if (active) {
      const float* bsp = &Bs[bf][0];
#pragma unroll 4
      for (int kk = 0; kk < KP; kk += 4) {
        v2f a[4], b[2];
#pragma unroll
        for (int mi = 0; mi < 4; ++mi) {
          if (AVEC) {
            a[mi] = gload_v2f(pa[mi]);                 // one global b64 load
          } else {
            a[mi].x = gload_f(pa[mi]);
            a[mi].y = gload_f(pa[mi] + sAk);
          }
        }
        __builtin_prefetch(pa[0] + apf, 0, 3);
#pragma unroll
        for (int ni = 0; ni < 2; ++ni)                 // ds_load_b64
          b[ni] = *(const v2f*)&bsp[bco[ni] * LDSP + kk + 2 * half];
#pragma unroll
        for (int mi = 0; mi < 4; ++mi)
#pragma unroll
          for (int ni = 0; ni < 2; ++ni)
            acc[mi][ni] = __builtin_amdgcn_wmma_f32_16x16x4_f32(
                false, a[mi], false, b[ni], (short)0, acc[mi][ni], false, false);
#pragma unroll
        for (int mi = 0; mi < 4; ++mi) pa[mi] += astep;
      }
    }
    __syncthreads();                        // all reads of Bs[bf] done
    bf ^= 1;
  }

  if (active) {
#pragma unroll
    for (int mi = 0; mi < 4; ++mi) {
#pragma unroll
      for (int ni = 0; ni < 2; ++ni) {
#pragma unroll
        for (int r = 0; r < 8; ++r) {
          const int row = row0 + mi * 16 + r + 8 * half;
          float x = acc[mi][ni][r] * alpha;
          if (hasR) x += gload_f(&R[(long)row * sRm + bc[ni]]);
          if (relu) x = fmaxf(x, 0.f);
          gstore_f(&D[(long)row * sDm + bc[ni]], x);
        }
      }
    }
  }
}

// ---------------------------------------------------------------------------
// Column softmax (token attention: softmax over key axis i).
// X: 32 batches of 256x256 (i-major). One block per z; thread = column j.
// ---------------------------------------------------------------------------
__global__ __launch_bounds__(256) void softmax_col(float* __restrict__ X)
{
  float* p = X + (long)blockIdx.x * (LTOK * LTOK) + threadIdx.x;
  float m = -3.402823466e38f;
  for (int i = 0; i < LTOK; ++i) m = fmaxf(m, gload_f(p + (long)i * LTOK));
  float s = 0.f;
  for (int i = 0; i < LTOK; ++i) s += __expf(gload_f(p + (long)i * LTOK) - m);
  const float inv = 1.f / s;
  for (int i = 0; i < LTOK; ++i) {
    const long o = (long)i * LTOK;
    gstore_f(p + o, __expf(gload_f(p + o) - m) * inv);
  }
}

// ---------------------------------------------------------------------------
// Row softmax over L=256 for cross-attn scores.
// raw layout: (z=32, 4096, 256); reads row (z, s0+blockIdx.x);
// writes coef chunk (z, NCHUNK, 256).
// ---------------------------------------------------------------------------
__global__ __launch_bounds__(256) void softmax_row(
    const float* __restrict__ raw, float* __restrict__ coef, int s0)
{
  const int z = blockIdx.y, s = blockIdx.x, t = threadIdx.x;
  const float x = gload_f(raw + ((long)z * NPTS + s0 + s) * LTOK + t);
  __shared__ float red[256];
  red[t] = x; __syncthreads();
  for (int off = 128; off > 0; off >>= 1) {
    if (t < off) red[t] = fmaxf(red[t], red[t + off]);
    __syncthreads();
  }
  const float m = red[0]; __syncthreads();
  const float e = __expf(x - m);
  red[t] = e; __syncthreads();
  for (int off = 128; off > 0; off >>= 1) {
    if (t < off) red[t] += red[t + off];
    __syncthreads();
  }
  gstore_f(coef + ((long)z * NCHUNK + s) * LTOK + t, e / red[0]);
}

// ---------------------------------------------------------------------------
static void gemm(hipStream_t st, const float* A, const float* B,
                 const float* R, float* D, int M, int N, int K,
                 long sAm, long sAk, long sAb,
                 long sBk, long sBn, long sBb,
                 long sRm, long sRb, long sDm, long sDb,
                 float alpha, int relu, int nb)
{
  dim3 grid(N / 128, (M + 127) / 128, nb);
  if (sAk == 1)
    wmma_gemm_f32<true><<<grid, 256, 0, st>>>(A, B, R, D, M, N, K,
        sAm, sAk, sAb, sBk, sBn, sBb, sRm, sRb, sDm, sDb,
        alpha, relu, R != nullptr);
  else
    wmma_gemm_f32<false><<<grid, 256, 0, st>>>(A, B, R, D, M, N, K,
        sAm, sAk, sAb, sBk, sBn, sBb, sRm, sRb, sDm, sDb,
        alpha, relu, R != nullptr);
}

extern "C" void kernel_launch(void* const* d_in, const int* in_sizes, int n_in,
                              void* d_out, int out_size, void* d_ws, size_t ws_size,
                              hipStream_t stream)
{
  (void)in_sizes; (void)n_in; (void)out_size; (void)ws_size;
  const float* in_feat  = (const float*)d_in[0];   // (8,768,4096)
  const float* in_tok   = (const float*)d_in[1];   // (8,512,256)
  const int*   knn      = (const int*)d_in[2];     // (8,256,16)
  const float* W_dyn1   = (const float*)d_in[3];   // (512,768)
  const float* W_dyn2   = (const float*)d_in[4];   // (512,512)
  const float* W_k      = (const float*)d_in[5];   // (256,512)
  const float* W_q      = (const float*)d_in[6];   // (256,512)
  const float* W_v      = (const float*)d_in[7];   // (512,512)
  const float* W_tff1   = (const float*)d_in[8];   // (1024,512)
  const float* W_tff2   = (const float*)d_in[9];   // (512,1024)
  const float* W_pv     = (const float*)d_in[10];  // (768,512)
  const float* W_pk     = (const float*)d_in[11];  // (768,512)
  const float* W_pq     = (const float*)d_in[12];  // (768,768)
  const float* W_pff1   = (const float*)d_in[13];  // (1536,768)
  const float* W_pff2   = (const float*)d_in[14];  // (768,1536)

  float* out      = (float*)d_out;                              // (8,768,4096)
  float* tok_out  = out + (long)BATCH * OC * NPTS;              // (8,512,256)
  float* raw_out  = tok_out + (long)BATCH * TC * LTOK;          // (8,4,4096,256)

  float* ws = (float*)d_ws;
  // early-phase buffers (dead before the projector chunk phase reuses [0,16M))
  float* pooled = ws + 0;          // 8*768*256  = 1572864
  float* mid    = ws + 1572864;    // 8*512*256  = 1048576
  float* tok    = ws + 2621440;    // 8*512*256  = 1048576
  float* kbuf   = ws + 3670016;    // 8*256*256  =  524288
  float* qbuf   = ws + 4194304;    // 8*256*256  =  524288
  float* vbuf   = ws + 4718592;    // 8*512*256  = 1048576
  float* kq     = ws + 5767168;    // 32*256*256 = 2097152
  float* xt     = ws + 7864320;    // 8*512*256  = 1048576
  float* t1     = ws + 8912896;    // 8*1024*256 = 2097152
  // projector chunk region (overlaps early buffers; temporally disjoint)
  float* pq_c   = ws + 0;          // 8*768*512   = 3145728
  float* coef_c = ws + 3145728;    // 32*512*256  = 4194304
  float* y_c    = ws + 7340032;    // 8*768*512   = 3145728
  float* f1_c   = ws + 10485760;   // 8*1536*512  = 6291456  -> end 16777216
  // persistent projector buffers
  float* pvb    = ws + 16777216;   // 8*768*256  = 1572864
  float* pkb    = ws + 18350080;   // 8*768*256  = 1572864  -> 19922944 floats

  // 1) gather + maxpool
  gather_maxpool<<<dim3(LTOK, BATCH), 256, 0, stream>>>(in_feat, knn, pooled);

  // 2) mid = relu(W_dyn1 @ pooled)            M=512 N=256 K=768
  gemm(stream, W_dyn1, pooled, nullptr, mid, TC, LTOK, CIN,
       CIN, 1, 0,  LTOK, 1, (long)CIN * LTOK,  0, 0,  LTOK, (long)TC * LTOK,
       1.f, 1, BATCH);
  // 3) tok = W_dyn2 @ mid + in_tokens         M=512 N=256 K=512
  gemm(stream, W_dyn2, mid, in_tok, tok, TC, LTOK, TC,
       TC, 1, 0,  LTOK, 1, (long)TC * LTOK,  LTOK, (long)TC * LTOK,
       LTOK, (long)TC * LTOK, 1.f, 0, BATCH);
  // 4/5/6) k,q,v projections
  gemm(stream, W_k, tok, nullptr, kbuf, TC / 2, LTOK, TC,
       TC, 1, 0,  LTOK, 1, (long)TC * LTOK,  0, 0,
       LTOK, (long)(TC / 2) * LTOK, 1.f, 0, BATCH);
  gemm(stream, W_q, tok, nullptr, qbuf, TC / 2, LTOK, TC,
       TC, 1, 0,  LTOK, 1, (long)TC * LTOK,  0, 0,
       LTOK, (long)(TC / 2) * LTOK, 1.f, 0, BATCH);
  gemm(stream, W_v, tok, nullptr, vbuf, TC, LTOK, TC,
       TC, 1, 0,  LTOK, 1, (long)TC * LTOK,  0, 0,
       LTOK, (long)TC * LTOK, 1.f, 0, BATCH);

  // 7) kq[z,i,j] = (1/8) * sum_c k[z,c,i]*q[z,c,j]   (A transposed), z=(b,h)
  gemm(stream, kbuf, qbuf, nullptr, kq, LTOK, LTOK, 64,
       1, LTOK, 64L * LTOK,  LTOK, 1, 64L * LTOK,  0, 0,
       LTOK, (long)LTOK * LTOK, 0.125f, 0, BATCH * HEADS);
  // 8) softmax over i (columns), in place
  softmax_col<<<BATCH * HEADS, 256, 0, stream>>>(kq);
  // 9) x_t = tok + v @ kq                     M=128 N=256 K=256 per (b,h)
  gemm(stream, vbuf, kq, tok, xt, TC / HEADS, LTOK, LTOK,
       LTOK, 1, (long)(TC / HEADS) * LTOK,
       LTOK, 1, (long)LTOK * LTOK,
       LTOK, (long)(TC / HEADS) * LTOK,
       LTOK, (long)(TC / HEADS) * LTOK, 1.f, 0, BATCH * HEADS);
  // 10) t1 = relu(W_tff1 @ x_t)               M=1024 N=256 K=512
  gemm(stream, W_tff1, xt, nullptr, t1, 2 * TC, LTOK, TC,
       TC, 1, 0,  LTOK, 1, (long)TC * LTOK,  0, 0,
       LTOK, 2L * TC * LTOK, 1.f, 1, BATCH);
  // 11) tokens_out = x_t + W_tff2 @ t1  (written to d_out)
  gemm(stream, W_tff2, t1, xt, tok_out, TC, LTOK, 2 * TC,
       2 * TC, 1, 0,  LTOK, 1, 2L * TC * LTOK,
       LTOK, (long)TC * LTOK,  LTOK, (long)TC * LTOK, 1.f, 0, BATCH);

  // 12) proj_v / proj_k from tokens_out
  gemm(stream, W_pv, tok_out, nullptr, pvb, OC, LTOK, TC,
       TC, 1, 0,  LTOK, 1, (long)TC * LTOK,  0, 0,
       LTOK, (long)OC * LTOK, 1.f, 0, BATCH);
  gemm(stream, W_pk, tok_out, nullptr, pkb, OC, LTOK, TC,
       TC, 1, 0,  LTOK, 1, (long)TC * LTOK,  0, 0,
       LTOK, (long)OC * LTOK, 1.f, 0, BATCH);

  const int hd = OC / HEADS;  // 192
  const float iscale = 0.07216878364870322f;  // 1/sqrt(192)

  // projector phase, chunked over N
  for (int s0 = 0; s0 < NPTS; s0 += NCHUNK) {
    // 13) pq_c = W_pq @ in_feat[:, :, s0:s0+NCHUNK]   M=768 N=512 K=768
    gemm(stream, W_pq, in_feat + s0, nullptr, pq_c, OC, NCHUNK, CIN,
         CIN, 1, 0,  NPTS, 1, (long)CIN * NPTS,  0, 0,
         NCHUNK, (long)OC * NCHUNK, 1.f, 0, BATCH);
    // 14) raw[z, s0+s, l] = iscale * sum_c pq_c[z,c,s]*pk[z,c,l]  (A^T)
    gemm(stream, pq_c, pkb, nullptr, raw_out + (long)s0 * LTOK,
         NCHUNK, LTOK, hd,
         1, NCHUNK, (long)hd * NCHUNK,
         LTOK, 1, (long)hd * LTOK,
         0, 0,
         LTOK, (long)NPTS * LTOK, iscale, 0, BATCH * HEADS);
    // 15) coef_c = softmax_l(raw chunk)
    softmax_row<<<dim3(NCHUNK, BATCH * HEADS), 256, 0, stream>>>(
        raw_out, coef_c, s0);
    // 16) y_c = in_feat_chunk + pv @ coef^T   M=192 N=512 K=256 per z (B^T)
    gemm(stream, pvb, coef_c, in_feat + s0, y_c, hd, NCHUNK, LTOK,
         LTOK, 1, (long)hd * LTOK,
         1, LTOK, (long)NCHUNK * LTOK,
         NPTS, (long)hd * NPTS,
         NCHUNK, (long)hd * NCHUNK, 1.f, 0, BATCH * HEADS);
    // 17) f1_c = relu(W_pff1 @ y_c)           M=1536 N=512 K=768
    gemm(stream, W_pff1, y_c, nullptr, f1_c, 2 * OC, NCHUNK, OC,
         OC, 1, 0,  NCHUNK, 1, (long)OC * NCHUNK,  0, 0,
         NCHUNK, 2L * OC * NCHUNK, 1.f, 1, BATCH);
    // 18) out_chunk = in_feat_chunk + W_pff2 @ f1_c
    gemm(stream, W_pff2, f1_c, in_feat + s0, out + s0, OC, NCHUNK, 2 * OC,
         2 * OC, 1, 0,  NCHUNK, 1, 2L * OC * NCHUNK,
         NPTS, (long)OC * NPTS,  NPTS, (long)OC * NPTS, 1.f, 0, BATCH);
  }
}